// Encoder_40166534152781
// MI455X (gfx1250) — compile-verified
//
#include <hip/hip_runtime.h>
#include <math.h>
#include <stdint.h>

#define NNODES 50000
#define NGRAPH 32
#define NEDGE  150000
#define NB_TOT (NGRAPH * NNODES)      // 1,600,000 nodes total
#define HDIM   6
#define KBIG   (NNODES * HDIM)        // 300,000
#define BN_EPS 1e-5f
#define SPLITK 100                    // K-splits for the big GEMM
#define KC     8                      // K-chunk staged in LDS per iteration

typedef float v2f __attribute__((ext_vector_type(2)));
typedef float v8f __attribute__((ext_vector_type(8)));

__device__ __forceinline__ v8f wmma_f32_16x16x4(v2f a, v2f b, v8f c) {
    // 8 args: (neg_a, A, neg_b, B, c_mod, C, reuse_a, reuse_b)
    return __builtin_amdgcn_wmma_f32_16x16x4_f32(false, a, false, b, (short)0, c,
                                                 false, false);
}

// CDNA5 async copy global -> LDS (ASYNCcnt-tracked, VGPR-bypassing)
__device__ __forceinline__ void async_load_b128(const float* gaddr, uint32_t lds_off) {
    asm volatile("global_load_async_to_lds_b128 %0, %1, off"
                 :: "v"(lds_off), "v"((unsigned long long)(uintptr_t)gaddr)
                 : "memory");
}
__device__ __forceinline__ void wait_asynccnt0() {
    asm volatile("s_wait_asynccnt 0x0" ::: "memory");
}

// ---------------------------------------------------------------- utilities
__global__ void k_zero(float* p, long long n) {
    long long i = (long long)blockIdx.x * blockDim.x + threadIdx.x;
    if (i < n) p[i] = 0.f;
}

// deg over the per-graph template (edges only; self-loop added analytically)
__global__ void k_count_deg(const int* __restrict__ eb, float* __restrict__ deg) {
    int e = blockIdx.x * blockDim.x + threadIdx.x;
    if (e < NEDGE) atomicAdd(&deg[eb[NEDGE + e]], 1.f);
}

__global__ void k_dis(const float* __restrict__ deg, float* __restrict__ dis,
                      float* __restrict__ lnorm) {
    int n = blockIdx.x * blockDim.x + threadIdx.x;
    if (n < NNODES) {
        float r = rsqrtf(deg[n] + 1.f);   // +1 self loop; always > 0
        dis[n] = r;
        lnorm[n] = r * r;                 // self-loop norm = dis^2
    }
}

__global__ void k_edge_norm(const int* __restrict__ eb, const float* __restrict__ dis,
                            float* __restrict__ norm_e) {
    int e = blockIdx.x * blockDim.x + threadIdx.x;
    if (e < NEDGE) norm_e[e] = dis[eb[e]] * dis[eb[NEDGE + e]];
}

// ---------------------------------------------------------------- GCN layer
// Q[i][0..5] = P[i][0..cin-1] @ W[cin][6]   (bias omitted: cancels in BN)
__global__ void k_transform(const float* __restrict__ P, const float* __restrict__ W,
                            float* __restrict__ Q, int cin) {
    long long i = (long long)blockIdx.x * blockDim.x + threadIdx.x;
    if (i >= NB_TOT) return;
    float h[HDIM];
    for (int c = 0; c < cin; ++c) h[c] = P[i * cin + c];
    for (int o = 0; o < HDIM; ++o) {
        float s = 0.f;
        for (int c = 0; c < cin; ++c) s += h[c] * W[c * HDIM + o];
        Q[i * HDIM + o] = s;
    }
}

// A[i][c] = Q[i][c] * lnorm[i % N]  (self-loop term doubles as the zero-init)
__global__ void k_init_agg(const float* __restrict__ Q, const float* __restrict__ lnorm,
                           float* __restrict__ A, float* __restrict__ stats) {
    long long i = (long long)blockIdx.x * blockDim.x + threadIdx.x;
    if (i < 16) stats[i] = 0.f;          // reset BN accumulators (stream-ordered)
    if (i >= NB_TOT) return;
    float w = lnorm[(int)(i % NNODES)];
    #pragma unroll
    for (int c = 0; c < HDIM; ++c) A[i * HDIM + c] = Q[i * HDIM + c] * w;
}

// scatter-add over B*E batched edges (template edges offset per graph)
__global__ void k_scatter(const int* __restrict__ eb, const float* __restrict__ norm_e,
                          const float* __restrict__ Q, float* __restrict__ A) {
    long long t = (long long)blockIdx.x * blockDim.x + threadIdx.x;
    if (t >= (long long)NGRAPH * NEDGE) return;
    int e = (int)(t % NEDGE);
    int g = (int)(t / NEDGE);
    long long row = (long long)eb[e]         + (long long)g * NNODES;
    long long col = (long long)eb[NEDGE + e] + (long long)g * NNODES;
    float w = norm_e[e];
    const float* src = Q + row * HDIM;
    float*       dst = A + col * HDIM;
    #pragma unroll
    for (int c = 0; c < HDIM; ++c) atomicAdd(&dst[c], src[c] * w);
}

// per-channel sum / sumsq over all 1.6M rows -> stats[0..5]=sum, [6..11]=sumsq
__global__ void k_bn_stats(const float* __restrict__ A, float* __restrict__ stats) {
    __shared__ float red[256];
    float s[HDIM] = {0}, q[HDIM] = {0};
    long long stride = (long long)gridDim.x * blockDim.x;
    for (long long i = (long long)blockIdx.x * blockDim.x + threadIdx.x; i < NB_TOT;
         i += stride) {
        #pragma unroll
        for (int c = 0; c < HDIM; ++c) {
            float v = A[i * HDIM + c];
            s[c] += v;
            q[c] += v * v;
        }
    }
    for (int k = 0; k < 2 * HDIM; ++k) {
        red[threadIdx.x] = (k < HDIM) ? s[k] : q[k - HDIM];
        __syncthreads();
        for (int off = 128; off > 0; off >>= 1) {
            if (threadIdx.x < off) red[threadIdx.x] += red[threadIdx.x + off];
            __syncthreads();
        }
        if (threadIdx.x == 0) atomicAdd(&stats[k], red[0]);
        __syncthreads();
    }
}

__global__ void k_bn_apply(float* __restrict__ A, const float* __restrict__ stats,
                           const float* __restrict__ g, const float* __restrict__ be) {
    long long i = (long long)blockIdx.x * blockDim.x + threadIdx.x;
    if (i >= (long long)NB_TOT * HDIM) return;
    int c = (int)(i % HDIM);
    const float inv = 1.f / (float)NB_TOT;
    float mu  = stats[c] * inv;
    float var = stats[HDIM + c] * inv - mu * mu;
    float v = (A[i] - mu) * rsqrtf(var + BN_EPS) * g[c] + be[c];
    A[i] = v > 0.f ? v : 0.f;
}

// ---------------------------------------------------------------- MLP (WMMA)
// t1[32,256] += A[32,300000] @ Bw[300000,256].
// One 16-wave block per K-split; operands staged via async global->LDS copies.
// Each wave owns one 16-col tile and computes BOTH 16-row tiles (B reuse),
// so lw1 (307 MB) is streamed exactly once and A exactly once.
__global__ void __launch_bounds__(512)
k_mlp1_wmma(const float* __restrict__ A, const float* __restrict__ Bw,
            float* __restrict__ C) {
    __shared__ float Bs[KC * 256];   // 8 KB: B rows kk..kk+7, all 256 cols
    __shared__ float As[32 * KC];    // 1 KB: A rows 0..31, cols kk..kk+7
    const int Nn = 256;
    const int t    = threadIdx.x;
    const int wave = t >> 5;          // N-tile index (0..15)
    const int lane = t & 31;
    const int ln   = lane & 15, half = lane >> 4;

    const int kspan = KBIG / SPLITK;  // 3000 (divisible by KC)
    const int k0 = blockIdx.x * kspan;

    const uint32_t bsb = (uint32_t)(uintptr_t)(void*)Bs; // LDS byte offset
    const uint32_t asb = (uint32_t)(uintptr_t)(void*)As;

    // async-copy assignments: 512 x b128 covers the 8 KB B panel exactly
    const int br = t >> 6;            // B row in chunk  (0..7)
    const int bc = (t & 63) * 4;      // B col (float)   16B granules
    const int ar = t >> 1;            // A row           (0..31), threads 0..63
    const int ac = (t & 1) * 4;       // A col (float)

    v8f acc0 = {0, 0, 0, 0, 0, 0, 0, 0};
    v8f acc1 = {0, 0, 0, 0, 0, 0, 0, 0};
    const int ncol = wave * 16 + ln;

    for (int kk = k0; kk < k0 + kspan; kk += KC) {
        async_load_b128(Bw + (long long)(kk + br) * Nn + bc,
                        bsb + (uint32_t)(br * Nn + bc) * 4u);
        if (t < 64)
            async_load_b128(A + (long long)ar * KBIG + kk + ac,
                            asb + (uint32_t)(ar * KC + ac) * 4u);
        wait_asynccnt0();
        __syncthreads();
        #pragma unroll
        for (int s = 0; s < KC; s += 4) {
            int ka = s + 2 * half;               // K layout per ISA 7.12.2
            v2f b, a0, a1;
            b.x  = Bs[ka * Nn + ncol];
            b.y  = Bs[(ka + 1) * Nn + ncol];
            a0.x = As[ln * KC + ka];
            a0.y = As[ln * KC + ka + 1];
            a1.x = As[(16 + ln) * KC + ka];
            a1.y = As[(16 + ln) * KC + ka + 1];
            acc0 = wmma_f32_16x16x4(a0, b, acc0);  // M-tile 0
            acc1 = wmma_f32_16x16x4(a1, b, acc1);  // M-tile 1 (B reused)
        }
        __syncthreads();
    }
    #pragma unroll
    for (int v = 0; v < 8; ++v) {                 // C layout: M = v + 8*half
        atomicAdd(&C[(v + 8 * half) * Nn + ncol],        acc0[v]);
        atomicAdd(&C[(16 + v + 8 * half) * Nn + ncol],   acc1[v]);
    }
}

__global__ void k_bias_relu(float* __restrict__ t, const float* __restrict__ b, int Nn,
                            int total) {
    int i = blockIdx.x * blockDim.x + threadIdx.x;
    if (i >= total) return;
    float v = t[i] + b[i % Nn];
    t[i] = v > 0.f ? v : 0.f;
}

// small GEMM: C[32,Nn] = act(A[32,K] @ Bw[K,Nn] + bias); one wave per 16x16 tile
__global__ void k_mlp_small(const float* __restrict__ A, const float* __restrict__ Bw,
                            const float* __restrict__ bias, float* __restrict__ C,
                            int K, int Nn, int do_relu) {
    int nt = blockIdx.x % (Nn / 16);
    int mt = blockIdx.x / (Nn / 16);
    int ln = threadIdx.x & 15, half = threadIdx.x >> 4;
    v8f acc = {0, 0, 0, 0, 0, 0, 0, 0};
    int mrow = mt * 16 + ln;
    int ncol = nt * 16 + ln;
    for (int kk = 0; kk < K; kk += 4) {
        int ka = kk + 2 * half;
        v2f a, b;
        a.x = A[mrow * K + ka];
        a.y = A[mrow * K + ka + 1];
        b.x = Bw[ka * Nn + ncol];
        b.y = Bw[(ka + 1) * Nn + ncol];
        if (kk + 36 < K)
            __builtin_prefetch(&Bw[(ka + 32) * Nn + ncol], 0, 1);
        acc = wmma_f32_16x16x4(a, b, acc);
    }
    #pragma unroll
    for (int v = 0; v < 8; ++v) {
        int r = mt * 16 + v + 8 * half;
        float val = acc[v] + bias[ncol];
        if (do_relu) val = val > 0.f ? val : 0.f;
        C[r * Nn + ncol] = val;
    }
}

// ---------------------------------------------------------------- launch
extern "C" void kernel_launch(void* const* d_in, const int* in_sizes, int n_in,
                              void* d_out, int out_size, void* d_ws, size_t ws_size,
                              hipStream_t stream) {
    const float* x   = (const float*)d_in[0];
    const int*   eb  = (const int*)d_in[1];
    const float* W1  = (const float*)d_in[2];
    const float* g1  = (const float*)d_in[4];
    const float* be1 = (const float*)d_in[5];
    const float* W2  = (const float*)d_in[6];
    const float* g2  = (const float*)d_in[8];
    const float* be2 = (const float*)d_in[9];
    const float* W3  = (const float*)d_in[10];
    const float* g3  = (const float*)d_in[12];
    const float* be3 = (const float*)d_in[13];
    const float* lw1 = (const float*)d_in[14];
    const float* lb1 = (const float*)d_in[15];
    const float* lw2 = (const float*)d_in[16];
    const float* lb2 = (const float*)d_in[17];
    const float* lw3 = (const float*)d_in[18];
    const float* lb3 = (const float*)d_in[19];
    const float* lw4 = (const float*)d_in[20];
    const float* lb4 = (const float*)d_in[21];

    // workspace carve-up (floats)
    float* ws     = (float*)d_ws;
    float* deg    = ws;                       // N
    float* dis    = deg    + NNODES;          // N
    float* lnorm  = dis    + NNODES;          // N
    float* norm_e = lnorm  + NNODES;          // E
    float* stats  = norm_e + NEDGE;           // 16
    float* t1     = stats  + 16;              // 32*256
    float* t2     = t1     + 32 * 256;        // 32*128
    float* t3     = t2     + 32 * 128;        // 32*64
    float* bufQ   = t3     + 32 * 64;         // NB*6
    float* bufA   = bufQ   + (long long)NB_TOT * HDIM; // NB*6

    const int TB = 256;
    // ---- GCN normalization (per-graph template, computed once)
    k_zero<<<(NNODES + TB - 1) / TB, TB, 0, stream>>>(deg, NNODES);
    k_count_deg<<<(NEDGE + TB - 1) / TB, TB, 0, stream>>>(eb, deg);
    k_dis<<<(NNODES + TB - 1) / TB, TB, 0, stream>>>(deg, dis, lnorm);
    k_edge_norm<<<(NEDGE + TB - 1) / TB, TB, 0, stream>>>(eb, dis, norm_e);

    // ---- 3x (GCNConv -> BN -> ReLU)
    const float* Ws[3]  = {W1, W2, W3};
    const float* gs[3]  = {g1, g2, g3};
    const float* bes[3] = {be1, be2, be3};
    const float* Pin = x;
    int cin = 3;
    const int gNB  = (NB_TOT + TB - 1) / TB;                           // 6250
    const int gE   = (int)(((long long)NGRAPH * NEDGE + TB - 1) / TB); // 18750
    const int gNBH = (int)(((long long)NB_TOT * HDIM + TB - 1) / TB);  // 37500
    for (int l = 0; l < 3; ++l) {
        k_transform<<<gNB, TB, 0, stream>>>(Pin, Ws[l], bufQ, cin);
        k_init_agg<<<gNB, TB, 0, stream>>>(bufQ, lnorm, bufA, stats);
        k_scatter<<<gE, TB, 0, stream>>>(eb, norm_e, bufQ, bufA);
        k_bn_stats<<<512, TB, 0, stream>>>(bufA, stats);
        k_bn_apply<<<gNBH, TB, 0, stream>>>(bufA, stats, gs[l], bes[l]);
        Pin = bufA;
        cin = HDIM;
    }

    // ---- MLP head: [32,300000] -> 256 -> 128 -> 64 -> 32
    k_zero<<<(32 * 256 + TB - 1) / TB, TB, 0, stream>>>(t1, 32 * 256);
    k_mlp1_wmma<<<SPLITK, 512, 0, stream>>>(bufA, lw1, t1);
    k_bias_relu<<<(32 * 256 + TB - 1) / TB, TB, 0, stream>>>(t1, lb1, 256, 32 * 256);
    k_mlp_small<<<2 * 8, 32, 0, stream>>>(t1, lw2, lb2, t2, 256, 128, 1);
    k_mlp_small<<<2 * 4, 32, 0, stream>>>(t2, lw3, lb3, t3, 128, 64, 1);
    k_mlp_small<<<2 * 2, 32, 0, stream>>>(t3, lw4, lb4, (float*)d_out, 64, 32, 0);
}